// ViewLearner_86440511800223
// MI455X (gfx1250) — compile-verified
//
#include <hip/hip_runtime.h>
#include <hip/hip_bf16.h>
#include <stdint.h>

typedef float v2f __attribute__((ext_vector_type(2)));
typedef float v8f __attribute__((ext_vector_type(8)));

#define N_EDGES          640000
#define DIMD             128
#define DIMH             64
#define TILES            (N_EDGES / 16)   // 40000 tiles of 16 edges
#define BLOCKS           800
#define WAVES            7                 // 224 threads; LDS = 64KB + 7*2*16.5KB = 302080 B < 320KB
#define ROW_PITCH        132               // dwords (128 + 4 pad -> conflict-free A reads)
#define ROW_BYTES        (ROW_PITCH * 4)   // 528
#define BUF_FLOATS       (32 * ROW_PITCH)  // 4224 floats per node buffer
#define BUF_BYTES        (32 * ROW_BYTES)  // 16896 B
#define W1_FLOATS        (256 * 64)        // 16384 floats = 64 KB, at LDS offset 0

__global__ __launch_bounds__(WAVES * 32, 1)
void edge_mlp_wmma_async(const float* __restrict__ node_emb,
                         const int*   __restrict__ edge_index,
                         const float* __restrict__ W1,
                         const float* __restrict__ b1,
                         const float* __restrict__ W2,
                         const float* __restrict__ b2,
                         float*       __restrict__ out)
{
    // single arena: W1 fragments first so every B-frag ds offset fits the 16-bit imm
    __shared__ __attribute__((aligned(16))) float smem[W1_FLOATS + WAVES * 2 * BUF_FLOATS];
    float* shW1 = smem;

    const int tid  = threadIdx.x;
    const int lane = tid & 31;
    const int wave = tid >> 5;
    const int hi   = lane >> 4;
    const int l15  = lane & 15;

    // ---- stage W1 into LDS in exact B-fragment layout (once per block) ----
    // B frag (4x16 f32): lane L holds pair B(k0+2*(L>>4), n), B(k0+2*(L>>4)+1, n), n = nt*16+(L&15)
    for (int slot = tid; slot < 8192; slot += WAVES * 32) {
        const int fi = slot >> 5;          // ks*4 + nt
        const int lp = slot & 31;
        const int ks = fi >> 2;
        const int nt = fi & 3;
        const int k  = ks * 4 + 2 * (lp >> 4);
        const int n  = nt * 16 + (lp & 15);
        shW1[slot * 2 + 0] = W1[(k + 0) * DIMH + n];
        shW1[slot * 2 + 1] = W1[(k + 1) * DIMH + n];
    }

    float b1r[4], w2r[4];
#pragma unroll
    for (int nt = 0; nt < 4; ++nt) {
        b1r[nt] = b1[nt * 16 + l15];
        w2r[nt] = W2[nt * 16 + l15];
    }
    const float b2v = b2[0];

    __syncthreads();   // only block-wide barrier: shW1 visible to all waves

    const uint32_t smemBase    = (uint32_t)(uintptr_t)smem;
    const uint32_t myNodesBase = smemBase + (uint32_t)(W1_FLOATS * 4) + (uint32_t)wave * (2u * BUF_BYTES);

    // stage one 16-edge tile (32 node rows, 512B each) into an LDS buffer via async DMA
    auto stage_async = [&](int tile, int buf) {
        const int nid = edge_index[hi * N_EDGES + tile * 16 + l15];   // lane L -> row L's node id
        const uint32_t ldsLane = myNodesBase + (uint32_t)buf * BUF_BYTES + (uint32_t)lane * 16u;
#pragma unroll
        for (int r = 0; r < 32; ++r) {
            const uint32_t rn   = (uint32_t)__shfl(nid, r, 32);
            const uint32_t goff = rn * (uint32_t)(DIMD * 4) + (uint32_t)lane * 16u;  // < 2^26
            const uint32_t la   = ldsLane + (uint32_t)(r * ROW_BYTES);
            asm volatile("global_load_async_to_lds_b128 %0, %1, %2"
                         :: "v"(la), "v"(goff), "s"(node_emb)
                         : "memory");
        }
    };

    // global-wave grid-stride over tiles; waves are fully independent (no in-loop barriers)
    const int gw     = blockIdx.x * WAVES + wave;
    const int stride = BLOCKS * WAVES;

    int buf = 0;
    stage_async(gw, 0);

    for (int t = gw; t < TILES; t += stride) {
        const int nxt = t + stride;
        if (nxt < TILES) {
            stage_async(nxt, buf ^ 1);                           // prefetch next tile
            // async loads complete in order: <=32 outstanding => this tile's 32 landed
            asm volatile("s_wait_asynccnt 0x20" ::: "memory");
        } else {
            asm volatile("s_wait_asynccnt 0x0" ::: "memory");
        }

        const float* nodeBuf = smem + W1_FLOATS + wave * (2 * BUF_FLOATS) + buf * BUF_FLOATS;
        const float* aS = nodeBuf + l15 * ROW_PITCH + 2 * hi;    // src rows (k < 128)
        const float* aD = aS + 16 * ROW_PITCH;                   // dst rows (k >= 128)
        const float* bB = shW1 + lane * 2;

        v8f acc[4];
#pragma unroll
        for (int nt = 0; nt < 4; ++nt) {
            v8f c;
#pragma unroll
            for (int j = 0; j < 8; ++j) c[j] = b1r[nt];          // C(m,n) = b1[n]
            acc[nt] = c;
        }

        // layer 1: [16 x 256] @ [256 x 64] via V_WMMA_F32_16X16X4_F32
#pragma unroll 8
        for (int ks = 0; ks < 32; ++ks) {
            const v2f a = *(const v2f*)(aS + ks * 4);
#pragma unroll
            for (int nt = 0; nt < 4; ++nt) {
                const v2f bf = *(const v2f*)(bB + (ks * 4 + nt) * 64);
                acc[nt] = __builtin_amdgcn_wmma_f32_16x16x4_f32(
                    false, a, false, bf, (short)0, acc[nt], false, false);
            }
        }
#pragma unroll 8
        for (int ks = 0; ks < 32; ++ks) {
            const v2f a = *(const v2f*)(aD + ks * 4);
#pragma unroll
            for (int nt = 0; nt < 4; ++nt) {
                const v2f bf = *(const v2f*)(bB + ((ks + 32) * 4 + nt) * 64);
                acc[nt] = __builtin_amdgcn_wmma_f32_16x16x4_f32(
                    false, a, false, bf, (short)0, acc[nt], false, false);
            }
        }

        // layer 2: logit[m] = sum_n relu(h[m,n]) * W2[n] + b2
        // C layout: vgpr v, lane L -> (m = v + 8*hi, n = nt*16 + l15)
        float logit = 0.0f;
#pragma unroll
        for (int v = 0; v < 8; ++v) {
            float s = 0.0f;
#pragma unroll
            for (int nt = 0; nt < 4; ++nt) {
                float hv = acc[nt][v];
                hv = hv > 0.0f ? hv : 0.0f;
                s += hv * w2r[nt];
            }
            s += __shfl_xor(s, 1, 32);
            s += __shfl_xor(s, 2, 32);
            s += __shfl_xor(s, 4, 32);
            s += __shfl_xor(s, 8, 32);
            if (l15 == v) logit = s;
        }
        if (l15 < 8) {
            const int m = l15 + 8 * hi;
            __builtin_nontemporal_store(logit + b2v, &out[t * 16 + m]);
        }
        buf ^= 1;
    }
}

extern "C" void kernel_launch(void* const* d_in, const int* in_sizes, int n_in,
                              void* d_out, int out_size, void* d_ws, size_t ws_size,
                              hipStream_t stream) {
    const float* node_emb   = (const float*)d_in[0];
    const int*   edge_index = (const int*)  d_in[1];   // JAX x64 disabled -> int32
    const float* W1         = (const float*)d_in[2];
    const float* b1         = (const float*)d_in[3];
    const float* W2         = (const float*)d_in[4];
    const float* b2         = (const float*)d_in[5];
    float*       out        = (float*)d_out;

    edge_mlp_wmma_async<<<BLOCKS, WAVES * 32, 0, stream>>>(node_emb, edge_index, W1, b1, W2, b2, out);
}